// MambaLayer_24240795419277
// MI455X (gfx1250) — compile-verified
//
#include <hip/hip_runtime.h>
#include <hip/hip_bf16.h>
#include <math.h>

// ---------------------------------------------------------------------------
// Mamba layer forward for MI455X (gfx1250, wave32).
//  - GEMMs: V_WMMA_F32_16X16X4_F32, 64x64 register tile per wave, strides as
//    template constants so fragment loads use immediate offsets (no pointer
//    arrays -> no VGPR spills), __launch_bounds__ to widen the VGPR budget.
//  - Selective scan: 3-pass chunked associative scan (16 chunks x 128 steps).
// ---------------------------------------------------------------------------

#define D_MODEL 1024
#define D_STATE 16
#define D_CONV 4
#define D_INNER 2048
#define DT_RANK 64
#define SEQ_L 2048

#define NCHUNK 16
#define CHUNK_L (SEQ_L / NCHUNK)          // 128
#define DN (D_INNER * D_STATE)            // 32768 (d,n) pairs

typedef float v2f __attribute__((ext_vector_type(2)));
typedef float v8f __attribute__((ext_vector_type(8)));

// ---------------------------------------------------------------------------
// C[M,N] = act( A[M,K](row-major, LDA) * W[N,K]^T (row-major, LDW) + bias )
// One wave computes a (BM*16) x (BN*16) tile; per K-step of 4 it issues
// BM+BN contiguous float2 fragment loads and BM*BN v_wmma_f32_16x16x4_f32.
//
// f32 16x16x4 WMMA per-lane layout (ISA 7.12.2):
//   A: lane = 16*hi + m, vgpr v holds A[m][2*hi + v]
//   B: lane = 16*hi + n, vgpr v holds B[2*hi + v][n]   (B = W^T)
//   D: vgpr r, lane = 16*hi + n holds D[r + 8*hi][n]
// => each lane's fragment is a contiguous float2 at (row*LD + k + 2*hi);
//    row blocks differ by 16*LD floats = compile-time immediate offsets.
// ---------------------------------------------------------------------------
template <int LDA, int LDW, int LDC, int BM, int BN, int ACT>
__global__ __launch_bounds__(256, 1)
void wmma_gemm_kernel(const float* __restrict__ A,
                      const float* __restrict__ W,
                      const float* __restrict__ bias,
                      float* __restrict__ C,
                      int M, int N, int K) {
    const int lane = threadIdx.x & 31;
    const int wave = threadIdx.x >> 5;
    const int tile = blockIdx.x * (blockDim.x >> 5) + wave;

    const int tilesN = N / (BN * 16);
    const int tm = tile / tilesN;            // wave-uniform
    const int tn = tile % tilesN;
    if (tm >= M / (BM * 16)) return;         // wave-uniform guard (EXEC all-1)

    const int hi = lane >> 4;
    const int lo = lane & 15;

    // Single base pointer per matrix; all row blocks reached via immediate
    // offsets i*16*LD (folded into the load instruction's IOFFSET).
    const float* Abase = A + (size_t)(tm * BM * 16 + lo) * LDA + 2 * hi;
    const float* Wbase = W + (size_t)(tn * BN * 16 + lo) * LDW + 2 * hi;

    v8f acc[BM][BN] = {};

    for (int k = 0; k < K; k += 4) {
        v2f a[BM], b[BN];
#pragma unroll
        for (int i = 0; i < BM; ++i)
            a[i] = *(const v2f*)(Abase + k + i * (16 * LDA));
#pragma unroll
        for (int j = 0; j < BN; ++j)
            b[j] = *(const v2f*)(Wbase + k + j * (16 * LDW));
#pragma unroll
        for (int i = 0; i < BM; ++i)
#pragma unroll
            for (int j = 0; j < BN; ++j)
                acc[i][j] = __builtin_amdgcn_wmma_f32_16x16x4_f32(
                    false, a[i], false, b[j], (short)0, acc[i][j], false, false);
    }

    float* Cbase = C + (size_t)(tm * BM * 16 + hi * 8) * LDC + tn * BN * 16 + lo;
#pragma unroll
    for (int j = 0; j < BN; ++j) {
        const float bv = (ACT == 1) ? bias[tn * BN * 16 + j * 16 + lo] : 0.0f;
#pragma unroll
        for (int i = 0; i < BM; ++i) {
#pragma unroll
            for (int r = 0; r < 8; ++r) {
                float v = acc[i][j][r];
                if (ACT == 1) {
                    v += bv;
                    v = (v > 20.0f) ? v : log1pf(__expf(v)); // softplus
                }
                Cbase[(size_t)(i * 16 + r) * LDC + j * 16] = v;
            }
        }
    }
}

// ---------------------------------------------------------------------------
// Causal depthwise conv1d (k=4) + bias + SiLU.  xi = xz[:, 0:D_INNER].
// ---------------------------------------------------------------------------
__global__ void conv_silu_kernel(const float* __restrict__ xz,
                                 const float* __restrict__ conv_w,
                                 const float* __restrict__ conv_b,
                                 float* __restrict__ xc) {
    const int idx = blockIdx.x * blockDim.x + threadIdx.x;
    if (idx >= SEQ_L * D_INNER) return;
    const int d = idx & (D_INNER - 1);
    const int l = idx >> 11;

    float s = conv_b[d];
#pragma unroll
    for (int j = 0; j < D_CONV; ++j) {
        const int ls = l - (D_CONV - 1) + j;
        if (ls >= 0)
            s += conv_w[d * D_CONV + j] * xz[(size_t)ls * (2 * D_INNER) + d];
    }
    xc[idx] = s / (1.0f + __expf(-s));
}

// ---------------------------------------------------------------------------
// Scan pass 1: per (chunk, d, n) from zero state:
//   aprod = prod_l exp(delta*A),  hend = chunk-local scan result.
// ---------------------------------------------------------------------------
__global__ void scan_pass1_kernel(const float* __restrict__ xdbl,
                                  const float* __restrict__ delta,
                                  const float* __restrict__ xc,
                                  const float* __restrict__ A_log,
                                  float* __restrict__ aprod,   // [NCHUNK][DN]
                                  float* __restrict__ hend) {  // [NCHUNK][DN]
    const int tid = blockIdx.x * blockDim.x + threadIdx.x;
    const int dn = tid & (DN - 1);
    const int chunk = tid >> 15;                               // DN = 2^15
    const int d = dn >> 4;
    const int n = dn & 15;

    const float Aval = -__expf(A_log[dn]);
    float h = 0.0f, ap = 1.0f;
    const int l0 = chunk * CHUNK_L;
    for (int l = l0; l < l0 + CHUNK_L; ++l) {
        const float dl = delta[(size_t)l * D_INNER + d];
        const float xv = xc[(size_t)l * D_INNER + d];
        const float Bv = xdbl[(size_t)l * 96 + DT_RANK + n];
        const float a = __expf(dl * Aval);
        h = a * h + (dl * xv) * Bv;
        ap *= a;
    }
    aprod[tid] = ap;
    hend[tid] = h;
}

// ---------------------------------------------------------------------------
// Scan pass 2: serial combine over the 16 chunks per (d,n).
// ---------------------------------------------------------------------------
__global__ void scan_pass2_kernel(const float* __restrict__ aprod,
                                  const float* __restrict__ hend,
                                  float* __restrict__ hinit) { // [NCHUNK][DN]
    const int dn = blockIdx.x * blockDim.x + threadIdx.x;
    float h = 0.0f;
#pragma unroll
    for (int c = 0; c < NCHUNK; ++c) {
        hinit[(size_t)c * DN + dn] = h;
        h = aprod[(size_t)c * DN + dn] * h + hend[(size_t)c * DN + dn];
    }
}

// ---------------------------------------------------------------------------
// Scan pass 3: recompute each chunk from its correct initial state, butterfly
// reduce h*C over 16 state lanes, fuse D*xc skip + SiLU(z) gating.
// ---------------------------------------------------------------------------
__global__ void scan_pass3_kernel(const float* __restrict__ xdbl,
                                  const float* __restrict__ delta,
                                  const float* __restrict__ xc,
                                  const float* __restrict__ xz,
                                  const float* __restrict__ A_log,
                                  const float* __restrict__ Dp_,
                                  const float* __restrict__ hinit,
                                  float* __restrict__ yg) {
    const int tid = blockIdx.x * blockDim.x + threadIdx.x;
    const int dn = tid & (DN - 1);
    const int chunk = tid >> 15;
    const int d = dn >> 4;
    const int n = dn & 15;

    const float Aval = -__expf(A_log[dn]);
    const float Dp = Dp_[d];
    float h = hinit[tid];

    const int l0 = chunk * CHUNK_L;
    for (int l = l0; l < l0 + CHUNK_L; ++l) {
        const float dl = delta[(size_t)l * D_INNER + d];
        const float xv = xc[(size_t)l * D_INNER + d];
        const float Bv = xdbl[(size_t)l * 96 + DT_RANK + n];
        const float Cv = xdbl[(size_t)l * 96 + DT_RANK + D_STATE + n];

        h = __expf(dl * Aval) * h + (dl * xv) * Bv;
        float acc = h * Cv;
        acc += __shfl_xor(acc, 1, 32);
        acc += __shfl_xor(acc, 2, 32);
        acc += __shfl_xor(acc, 4, 32);
        acc += __shfl_xor(acc, 8, 32);
        if (n == 0) {
            const float zv = xz[(size_t)l * (2 * D_INNER) + D_INNER + d];
            const float silu_z = zv / (1.0f + __expf(-zv));
            yg[(size_t)l * D_INNER + d] = (acc + Dp * xv) * silu_z;
        }
    }
}

// ---------------------------------------------------------------------------
// Host-side launcher
// ---------------------------------------------------------------------------
extern "C" void kernel_launch(void* const* d_in, const int* in_sizes, int n_in,
                              void* d_out, int out_size, void* d_ws, size_t ws_size,
                              hipStream_t stream) {
    const float* x         = (const float*)d_in[0];
    const float* in_proj_w = (const float*)d_in[1];
    const float* conv_w    = (const float*)d_in[2];
    const float* conv_b    = (const float*)d_in[3];
    const float* x_proj_w  = (const float*)d_in[4];
    const float* dt_proj_w = (const float*)d_in[5];
    const float* dt_proj_b = (const float*)d_in[6];
    const float* A_log     = (const float*)d_in[7];
    const float* D_param   = (const float*)d_in[8];
    const float* out_proj_w= (const float*)d_in[9];
    float* out = (float*)d_out;

    // workspace layout (floats)
    float* ws    = (float*)d_ws;
    float* xz    = ws;                                   // L * 4096
    float* xc    = xz    + (size_t)SEQ_L * 2 * D_INNER;  // L * 2048
    float* xdbl  = xc    + (size_t)SEQ_L * D_INNER;      // L * 96
    float* delta = xdbl  + (size_t)SEQ_L * 96;           // L * 2048
    float* yg    = delta + (size_t)SEQ_L * D_INNER;      // L * 2048
    float* aprod = yg    + (size_t)SEQ_L * D_INNER;      // NCHUNK * DN
    float* hend  = aprod + (size_t)NCHUNK * DN;          // NCHUNK * DN
    float* hinit = hend  + (size_t)NCHUNK * DN;          // NCHUNK * DN

    const int threads = 256;                 // 8 waves per block
    const int wpb = threads / 32;

    // 1) xz = x @ in_proj_w^T          M=2048 N=4096 K=1024   (64x64 tiles)
    {
        int tiles = (SEQ_L / 64) * ((2 * D_INNER) / 64);
        wmma_gemm_kernel<D_MODEL, D_MODEL, 2 * D_INNER, 4, 4, 0>
            <<<(tiles + wpb - 1) / wpb, threads, 0, stream>>>(
                x, in_proj_w, nullptr, xz, SEQ_L, 2 * D_INNER, D_MODEL);
    }

    // 2) xc = silu(causal_dwconv(xi) + b)
    {
        int total = SEQ_L * D_INNER;
        conv_silu_kernel<<<(total + threads - 1) / threads, threads, 0, stream>>>(
            xz, conv_w, conv_b, xc);
    }

    // 3) x_dbl = xc @ x_proj_w^T       M=2048 N=96 K=2048     (64x32 tiles)
    {
        int tiles = (SEQ_L / 64) * (96 / 32);
        wmma_gemm_kernel<D_INNER, D_INNER, 96, 4, 2, 0>
            <<<(tiles + wpb - 1) / wpb, threads, 0, stream>>>(
                xc, x_proj_w, nullptr, xdbl, SEQ_L, 96, D_INNER);
    }

    // 4) delta = softplus(dt_low @ dt_proj_w^T + b)  M=2048 N=2048 K=64 (64x64)
    {
        int tiles = (SEQ_L / 64) * (D_INNER / 64);
        wmma_gemm_kernel<96, DT_RANK, D_INNER, 4, 4, 1>
            <<<(tiles + wpb - 1) / wpb, threads, 0, stream>>>(
                xdbl, dt_proj_w, dt_proj_b, delta, SEQ_L, D_INNER, DT_RANK);
    }

    // 5) chunked associative scan
    {
        int total = NCHUNK * DN;             // 524288 threads
        scan_pass1_kernel<<<total / threads, threads, 0, stream>>>(
            xdbl, delta, xc, A_log, aprod, hend);
        scan_pass2_kernel<<<DN / threads, threads, 0, stream>>>(
            aprod, hend, hinit);
        scan_pass3_kernel<<<total / threads, threads, 0, stream>>>(
            xdbl, delta, xc, xz, A_log, D_param, hinit, yg);
    }

    // 6) out = yg @ out_proj_w^T       M=2048 N=1024 K=2048   (64x64 tiles)
    {
        int tiles = (SEQ_L / 64) * (D_MODEL / 64);
        wmma_gemm_kernel<D_INNER, D_INNER, D_MODEL, 4, 4, 0>
            <<<(tiles + wpb - 1) / wpb, threads, 0, stream>>>(
                yg, out_proj_w, nullptr, out, SEQ_L, D_MODEL, D_INNER);
    }
}